// RoutingLayer_81853486727841
// MI455X (gfx1250) — compile-verified
//
#include <hip/hip_runtime.h>
#include <math.h>

// RoutingLayer on MI455X (gfx1250, wave32, WMMA).
// All matmuls use V_WMMA_F32_16X16X4_F32 (exact fp32).
//
// Transposed tile formulation: per (k,j), D[z(16) x b(16)] = A(Wt 16x8) @ B(xt 8x16)
//   A frag (lane L: m=L&15=z, h=L>>4): VGPRv = W[k,j, i0+2h+v, m]
//   B frag (lane L: n=L&15=b, h=L>>4): VGPRv = x[b0+n, j, i0+2h+v]   (float2 load)
//   C/D  (lane L, VGPRv)             : element (z = v+8h, b = n)
// => dot over z = 8 in-lane FMAs + one shfl_xor(16); softmax over K is per-lane.

typedef __attribute__((ext_vector_type(2))) float v2f;
typedef __attribute__((ext_vector_type(4))) float v4f;
typedef __attribute__((ext_vector_type(8))) float v8f;

#define EPSQ 1e-20f

constexpr int B_ = 128, K_ = 32, J_ = 1152, I_ = 8, Z_ = 16;
constexpr int BT  = 16;          // batch tile (WMMA N)
constexpr int NBT = B_ / BT;     // 8 batch tiles
constexpr int JC  = 48;          // j per chunk in k_route
constexpr int NJC = J_ / JC;     // 24 chunks
constexpr int JW  = JC / 8;      // 6 j per wave in k_route
constexpr int JA  = J_ / 8;      // 144 j per wave in k_usum / k_agg

// workspace layout (float elements)
constexpr size_t WS_C_OFF = 0;                               // c_norm [J][NBT][K][16b]
constexpr size_t WS_C_SZ  = (size_t)J_ * NBT * K_ * 16;      // 4,718,592 floats
constexpr size_t WS_U_OFF = WS_C_SZ;                         // usum  [NBT][K][16z][16b]
constexpr size_t WS_U_SZ  = (size_t)NBT * K_ * 256;          // 65,536 floats
// total ws need: (WS_C_SZ + WS_U_SZ)*4 bytes ~= 18.3 MiB

__device__ __forceinline__ v8f wmma_f32(v2f a, v2f b, v8f c) {
    // (neg_a, A, neg_b, B, c_mod, C, reuse_a, reuse_b)
    return __builtin_amdgcn_wmma_f32_16x16x4_f32(false, a, false, b, (short)0, c,
                                                 false, false);
}

// ---------------------------------------------------------------------------
// Kernel 1: usum[b,k,z] = sum_{j,i} x[b,j,i] * W[k,j,i,z]   (transposed tiles)
// grid = NBT*K_ blocks, 256 threads (8 waves split J).
// ---------------------------------------------------------------------------
__global__ __launch_bounds__(256) void k_usum(const float* __restrict__ x,
                                              const float* __restrict__ W,
                                              float* __restrict__ ws) {
    __shared__ float red[8 * 256];
    const int bt   = blockIdx.x >> 5;
    const int k    = blockIdx.x & 31;
    const int w    = threadIdx.x >> 5;
    const int lane = threadIdx.x & 31;
    const int n = lane & 15, h = lane >> 4;
    const int b0 = bt * BT;

    const float* __restrict__ Wk = W + (size_t)k * J_ * I_ * Z_;
    const float* __restrict__ xr = x + (size_t)(b0 + n) * (J_ * I_);

    v8f acc0 = {}, acc1 = {};
    const int j0 = w * JA;
    for (int j = j0; j < j0 + JA; ++j) {
        v2f xb0 = *(const v2f*)(xr + j * 8 + 2 * h);        // i = 2h, 2h+1
        v2f xb1 = *(const v2f*)(xr + j * 8 + 4 + 2 * h);    // i = 4+2h, 5+2h
        const float* wb = Wk + (size_t)j * 128;
        v2f a0, a1;
        a0.x = wb[32 * h + n];        a0.y = wb[32 * h + 16 + n];
        a1.x = wb[64 + 32 * h + n];   a1.y = wb[80 + 32 * h + n];
        acc0 = wmma_f32(a0, xb0, acc0);   // two independent accumulation chains
        acc1 = wmma_f32(a1, xb1, acc1);
    }
#pragma unroll
    for (int v = 0; v < 8; ++v)
        red[w * 256 + (v + 8 * h) * 16 + n] = acc0[v] + acc1[v];
    __syncthreads();
    float s = 0.f;
#pragma unroll
    for (int ww = 0; ww < 8; ++ww) s += red[ww * 256 + threadIdx.x];  // fixed order
    float* usum = ws + WS_U_OFF;
    usum[((size_t)bt * K_ + k) * 256 + threadIdx.x] = s;   // [bt][k][z][b]
}

// ---------------------------------------------------------------------------
// Kernel 2: c[b,j,k] = softmax_k( dot_z(u, usum)/sqrt(Z) ) + bias[k,j]
// grid = NBT*NJC blocks, 256 threads (8 waves split the j-chunk, all 32 k each).
// ---------------------------------------------------------------------------
__global__ __launch_bounds__(256) void k_route(const float* __restrict__ x,
                                               const float* __restrict__ W,
                                               const float* __restrict__ bias,
                                               float* __restrict__ ws) {
    __shared__ float su[K_ * 256];   // usum in [k][h][b][v] layout -> float4 reads
    const int bt   = blockIdx.x / NJC;
    const int jc   = blockIdx.x % NJC;
    const int w    = threadIdx.x >> 5;
    const int lane = threadIdx.x & 31;
    const int n = lane & 15, h = lane >> 4;
    const int b0 = bt * BT;

    const float* __restrict__ usum = ws + WS_U_OFF;
    for (int idx = threadIdx.x; idx < K_ * 256; idx += 256) {
        int k = idx >> 8, r = idx & 255;
        int z = r >> 4, bb = r & 15;
        su[k * 256 + (z >> 3) * 128 + bb * 8 + (z & 7)] =
            usum[((size_t)bt * K_ + k) * 256 + r];
    }
    __syncthreads();

    const float* __restrict__ xr = x + (size_t)(b0 + n) * (J_ * I_);
    const int jbase = jc * JC + w * JW;
    for (int jj = 0; jj < JW; ++jj) {
        const int j = jbase + jj;
        v2f xb0 = *(const v2f*)(xr + j * 8 + 2 * h);
        v2f xb1 = *(const v2f*)(xr + j * 8 + 4 + 2 * h);
        float ck[K_];
#pragma unroll
        for (int k = 0; k < K_; ++k) {
            const float* wb = W + ((size_t)k * J_ + j) * 128;
            v2f a0, a1;
            a0.x = wb[32 * h + n];        a0.y = wb[32 * h + 16 + n];
            a1.x = wb[64 + 32 * h + n];   a1.y = wb[80 + 32 * h + n];
            v8f u = {};
            u = wmma_f32(a0, xb0, u);
            u = wmma_f32(a1, xb1, u);
            v4f s0 = *(const v4f*)(&su[k * 256 + h * 128 + n * 8]);
            v4f s1 = *(const v4f*)(&su[k * 256 + h * 128 + n * 8 + 4]);
            float p = u[0] * s0.x + u[1] * s0.y + u[2] * s0.z + u[3] * s0.w +
                      u[4] * s1.x + u[5] * s1.y + u[6] * s1.z + u[7] * s1.w;
            p += __shfl_xor(p, 16, 32);          // join the two z-halves
            ck[k] = p * 0.25f;                   // 1/sqrt(Z), Z=16
        }
        // softmax over K entirely in-lane (lane = batch)
        float mx = ck[0];
#pragma unroll
        for (int k = 1; k < K_; ++k) mx = fmaxf(mx, ck[k]);
        float sum = 0.f;
#pragma unroll
        for (int k = 0; k < K_; ++k) { ck[k] = __expf(ck[k] - mx); sum += ck[k]; }
        const float inv = 1.0f / sum;
        float* co = ws + WS_C_OFF + (size_t)j * (NBT * K_ * 16) + (size_t)bt * (K_ * 16);
#pragma unroll
        for (int k = 0; k < K_; ++k) {
            float wk = ck[k] * inv + bias[k * J_ + j];
            if (h == 0) co[k * 16 + n] = wk;     // one copy per (b, j, k)
        }
    }
}

// ---------------------------------------------------------------------------
// Kernel 3: s[b,k,z] = sum_j c[b,j,k] * u[b,k,j,z]; then squash; write out.
// grid = NBT*K_ blocks, 256 threads (8 waves split J).
// ---------------------------------------------------------------------------
__global__ __launch_bounds__(256) void k_agg(const float* __restrict__ x,
                                             const float* __restrict__ W,
                                             const float* __restrict__ ws,
                                             float* __restrict__ out) {
    __shared__ float red[8 * 256];
    const int bt   = blockIdx.x >> 5;
    const int k    = blockIdx.x & 31;
    const int w    = threadIdx.x >> 5;
    const int lane = threadIdx.x & 31;
    const int n = lane & 15, h = lane >> 4;
    const int b0 = bt * BT;

    const float* __restrict__ Wk = W + (size_t)k * J_ * 128;
    const float* __restrict__ xr = x + (size_t)(b0 + n) * (J_ * I_);
    const float* __restrict__ cb = ws + WS_C_OFF + (size_t)bt * (K_ * 16) + (size_t)k * 16 + n;

    v8f s = {};
    const int j0 = w * JA;
    for (int j = j0; j < j0 + JA; ++j) {
        v2f xb0 = *(const v2f*)(xr + j * 8 + 2 * h);
        v2f xb1 = *(const v2f*)(xr + j * 8 + 4 + 2 * h);
        const float* wb = Wk + (size_t)j * 128;
        v2f a0, a1;
        a0.x = wb[32 * h + n];        a0.y = wb[32 * h + 16 + n];
        a1.x = wb[64 + 32 * h + n];   a1.y = wb[80 + 32 * h + n];
        v8f u0 = {}, u1 = {};
        u0 = wmma_f32(a0, xb0, u0);
        u1 = wmma_f32(a1, xb1, u1);
        const float c = cb[(size_t)j * (NBT * K_ * 16)];   // c[b=n, j, k]
#pragma unroll
        for (int v = 0; v < 8; ++v) s[v] += c * (u0[v] + u1[v]);
    }
#pragma unroll
    for (int v = 0; v < 8; ++v)
        red[w * 256 + (v + 8 * h) * 16 + n] = s[v];
    __syncthreads();
    float sv = 0.f;
#pragma unroll
    for (int ww = 0; ww < 8; ++ww) sv += red[ww * 256 + threadIdx.x];  // fixed order
    __syncthreads();
    // remap to [b][z] for the per-(b) norm over z, then squash
    {
        int z = threadIdx.x >> 4, bb = threadIdx.x & 15;
        red[bb * 16 + z] = sv;
    }
    __syncthreads();
    const int b = threadIdx.x >> 4, z = threadIdx.x & 15;
    const float val = red[threadIdx.x];
    float ss = 0.f;
#pragma unroll
    for (int q = 0; q < 16; ++q) { float t = red[b * 16 + q]; ss += t * t; }
    const float nrm = sqrtf(ss);
    const float o = (1.0f - 1.0f / (__expf(nrm) + EPSQ)) * (val / (nrm + EPSQ));
    out[(size_t)(b0 + b) * (K_ * Z_) + k * Z_ + z] = o;   // [B][K][Z]
}

extern "C" void kernel_launch(void* const* d_in, const int* in_sizes, int n_in,
                              void* d_out, int out_size, void* d_ws, size_t ws_size,
                              hipStream_t stream) {
    (void)in_sizes; (void)n_in; (void)out_size; (void)ws_size;
    const float* x    = (const float*)d_in[0];   // [B, J, I]
    const float* W    = (const float*)d_in[1];   // [K, J, I, Z]
    const float* bias = (const float*)d_in[2];   // [K, J, 1]
    float* out = (float*)d_out;                  // [B, K, Z]
    float* ws  = (float*)d_ws;                   // needs ~18.3 MiB

    k_usum <<<dim3(NBT * K_),  dim3(256), 0, stream>>>(x, W, ws);
    k_route<<<dim3(NBT * NJC), dim3(256), 0, stream>>>(x, W, bias, ws);
    k_agg  <<<dim3(NBT * K_),  dim3(256), 0, stream>>>(x, W, ws, out);
}